// GramLayer_49435073577339
// MI455X (gfx1250) — compile-verified
//
#include <hip/hip_runtime.h>

typedef __attribute__((ext_vector_type(2))) float v2f;
typedef __attribute__((ext_vector_type(4))) float v4f;
typedef __attribute__((ext_vector_type(8))) float v8f;

#define PDIM 128

// ---------------------------------------------------------------------------
// Gauss-Jordan inverse of a 128x128 SPD matrix held in LDS (M), identity in I.
// On exit: I = M^-1, M destroyed. Returns sum(log(pivots)) == log(det M)
// (valid on thread 0 only). 1024 threads: thread t -> row = t&127, 32-column
// chunk q = t>>7 spanning the combined [M | I] 256 columns.
// ---------------------------------------------------------------------------
__device__ __forceinline__ float gj_invert_128(float* M, float* I, int tid) {
  float logdet = 0.0f;
  for (int p = 0; p < PDIM; ++p) {
    float piv = M[p * PDIM + p];
    if (tid == 0) logdet += logf(piv);
    __syncthreads();                       // everyone read pivot before scaling
    float inv = 1.0f / piv;
    if (tid < 2 * PDIM) {
      if (tid < PDIM) M[p * PDIM + tid] *= inv;
      else            I[p * PDIM + (tid - PDIM)] *= inv;
    }
    __syncthreads();
    int row = tid & (PDIM - 1);
    float f = M[row * PDIM + p];           // read factor before anyone writes col p
    __syncthreads();
    if (row != p) {
      int c0 = (tid >> 7) * 32;            // 0..224 over combined [M|I]
      if (c0 < PDIM) {
        #pragma unroll 8
        for (int j = 0; j < 32; ++j)
          M[row * PDIM + c0 + j] -= f * M[p * PDIM + c0 + j];
      } else {
        int cc = c0 - PDIM;
        #pragma unroll 8
        for (int j = 0; j < 32; ++j)
          I[row * PDIM + cc + j] -= f * I[p * PDIM + cc + j];
      }
    }
    __syncthreads();
  }
  return logdet;
}

// ---------------------------------------------------------------------------
// Setup kernel (single block): Kii_inv, G = V V^T/P * exp(s), logdets, trace,
// objective. Writes Kii_inv -> ws[0:P*P], G -> ws[P*P:2*P*P] and d_out blocks.
// ---------------------------------------------------------------------------
__global__ void __launch_bounds__(1024) setup_kernel(
    const float* __restrict__ K_ii, const float* __restrict__ V,
    const float* __restrict__ gscale, const int* __restrict__ dof,
    float* __restrict__ ws, float* __restrict__ out_Gii,
    float* __restrict__ out_obj) {
  __shared__ float sM[PDIM * PDIM];
  __shared__ float sI[PDIM * PDIM];
  __shared__ float sTrace;
  int tid = threadIdx.x;

  for (int i = tid; i < PDIM * PDIM; i += 1024) {
    sM[i] = K_ii[i];
    int r = i >> 7, c = i & (PDIM - 1);
    sI[i] = (r == c) ? 1.0f : 0.0f;
  }
  if (tid == 0) sTrace = 0.0f;
  __syncthreads();

  float logdetK = gj_invert_128(sM, sI, tid);   // sI = Kii_inv

  for (int i = tid; i < PDIM * PDIM; i += 1024) {
    ws[i] = sI[i];          // persist Kii_inv (symmetric)
    sM[i] = V[i];           // stage V for G = V V^T
  }
  __syncthreads();

  float scale = expf(gscale[0]) / (float)PDIM;
  float gv[16];
  float tr = 0.0f;
  #pragma unroll
  for (int u = 0; u < 16; ++u) {
    int idx = tid + 1024 * u;
    int r = idx >> 7, c = idx & (PDIM - 1);
    float s = 0.0f;
    for (int k = 0; k < PDIM; ++k) s += sM[r * PDIM + k] * sM[c * PDIM + k];
    s *= scale;
    gv[u] = s;
    tr += sI[idx] * s;      // trace(Kii_inv * G): both symmetric -> elementwise
  }
  atomicAdd(&sTrace, tr);
  __syncthreads();          // all reads of V done before overwriting sM

  #pragma unroll
  for (int u = 0; u < 16; ++u) {
    int idx = tid + 1024 * u;
    sM[idx] = gv[u];
    out_Gii[idx] = gv[u];
    ws[PDIM * PDIM + idx] = gv[u];
    int r = idx >> 7, c = idx & (PDIM - 1);
    sI[idx] = (r == c) ? 1.0f : 0.0f;
  }
  __syncthreads();

  float logdetG = gj_invert_128(sM, sI, tid);   // pivots of SPD G -> logdet G

  if (tid == 0) {
    float logdetKinv = -logdetK;
    float obj = -0.5f * (float)dof[0] *
                (-(logdetG + logdetKinv) + sTrace - (float)PDIM);
    out_obj[0] = obj;
  }
}

// ---------------------------------------------------------------------------
// Fused big kernel: per 16-row tile (one wave32 each):
//   A = ti @ Kii_inv      (256 x v_wmma_f32_16x16x4_f32)
//   ktt = K_t - rowsum(A*ti)
//   Gti = A @ G           (256 x v_wmma_f32_16x16x4_f32), stored to global
//   gt  = rowsum(Gti*A) + ktt
// Kii_inv and G are symmetric -> B-operand fragments are contiguous v2f loads
// from row-major LDS. Block = 256 threads = 8 waves = 128 rows.
// ---------------------------------------------------------------------------
__global__ void __launch_bounds__(256) fused_gram_kernel(
    const float* __restrict__ K_ti, const float* __restrict__ K_t,
    const float* __restrict__ ws,
    float* __restrict__ out_Gti, float* __restrict__ out_gt) {
  __shared__ float sKinv[PDIM * PDIM];      // 64 KB
  __shared__ float sG[PDIM * PDIM];         // 64 KB
  __shared__ float sA[8][16 * PDIM];        // 64 KB, per-wave A staging
  __shared__ float sDot[8][16];

  int tid = threadIdx.x;
  {                                         // cooperative vectorized fill
    const v4f* src = (const v4f*)ws;
    v4f* k4 = (v4f*)sKinv;
    v4f* g4 = (v4f*)sG;
    const int n4 = (PDIM * PDIM) / 4;
    for (int i = tid; i < n4; i += 256) {
      k4[i] = src[i];
      g4[i] = src[n4 + i];
    }
  }
  __syncthreads();

  const int w    = tid >> 5;
  const int lane = tid & 31;
  const int l16  = lane & 15;
  const int hk   = lane >> 4;               // half-wave: K pair select
  const long rowBase = (long)blockIdx.x * 128 + (long)w * 16;

  // A-operand fragments of ti: lane(l16,hk) holds (row=l16, cols 4k+2hk..+1)
  const float* tiRow = K_ti + (rowBase + l16) * PDIM + 2 * hk;
  v2f tiF[32];
  #pragma unroll
  for (int k = 0; k < 32; ++k) tiF[k] = *(const v2f*)(tiRow + 4 * k);

  float* sAw = sA[w];

  // ---- GEMM 1: A = ti @ Kinv ----
  #pragma unroll
  for (int cb = 0; cb < 8; ++cb) {
    v8f acc = {};
    const float* kp = &sKinv[(16 * cb + l16) * PDIM + 2 * hk];  // symmetric
    #pragma unroll
    for (int k = 0; k < 32; ++k) {
      v2f b = *(const v2f*)(kp + 4 * k);
      acc = __builtin_amdgcn_wmma_f32_16x16x4_f32(
          false, tiF[k], false, b, (short)0, acc, false, false);
    }
    // C-layout: vgpr r, lanes(l16,hk) -> (M = r + 8*hk, N = 16*cb + l16)
    float* ap = &sAw[(8 * hk) * PDIM + 16 * cb + l16];
    #pragma unroll
    for (int r = 0; r < 8; ++r) ap[r * PDIM] = acc[r];
  }
  __syncthreads();

  // Re-fragment A from LDS into A-operand layout
  const float* aRow = &sAw[l16 * PDIM + 2 * hk];
  v2f aF[32];
  #pragma unroll
  for (int k = 0; k < 32; ++k) aF[k] = *(const v2f*)(aRow + 4 * k);

  // rowsum(A * ti): per-lane partial over half the columns, pair via xor-16
  float d = 0.0f;
  #pragma unroll
  for (int k = 0; k < 32; ++k)
    d += tiF[k].x * aF[k].x + tiF[k].y * aF[k].y;
  d += __shfl_xor(d, 16, 32);
  if (lane < 16) sDot[w][lane] = d;         // full dot for row l16

  // ---- GEMM 2: Gti = A @ G, direct store + gt partials ----
  float gp[8];
  #pragma unroll
  for (int r = 0; r < 8; ++r) gp[r] = 0.0f;

  #pragma unroll
  for (int cb = 0; cb < 8; ++cb) {
    v8f acc = {};
    const float* gpp = &sG[(16 * cb + l16) * PDIM + 2 * hk];    // symmetric
    #pragma unroll
    for (int k = 0; k < 32; ++k) {
      v2f b = *(const v2f*)(gpp + 4 * k);
      acc = __builtin_amdgcn_wmma_f32_16x16x4_f32(
          false, aF[k], false, b, (short)0, acc, false, false);
    }
    float* go = out_Gti + (rowBase + 8 * hk) * PDIM + 16 * cb + l16;
    const float* av = &sAw[(8 * hk) * PDIM + 16 * cb + l16];
    #pragma unroll
    for (int r = 0; r < 8; ++r) {
      go[r * PDIM] = acc[r];                // 64B-contiguous per half-wave
      gp[r] += acc[r] * av[r * PDIM];
    }
  }

  // reduce gt partials across the 16 lanes of each half-wave
  #pragma unroll
  for (int m = 1; m < 16; m <<= 1) {
    #pragma unroll
    for (int r = 0; r < 8; ++r) gp[r] += __shfl_xor(gp[r], m, 32);
  }
  if (l16 == 0) {                           // lane 0 -> rows 0..7, lane 16 -> 8..15
    int rb = 8 * hk;
    #pragma unroll
    for (int r = 0; r < 8; ++r) {
      long row = rowBase + rb + r;
      out_gt[row] = gp[r] + K_t[row] - sDot[w][rb + r];
    }
  }
}

// ---------------------------------------------------------------------------
extern "C" void kernel_launch(void* const* d_in, const int* in_sizes, int n_in,
                              void* d_out, int out_size, void* d_ws,
                              size_t ws_size, hipStream_t stream) {
  const float* K_ii = (const float*)d_in[0];
  const float* K_ti = (const float*)d_in[1];
  const float* K_t  = (const float*)d_in[2];
  const float* V    = (const float*)d_in[3];
  const float* gsc  = (const float*)d_in[4];
  const int*   dof  = (const int*)d_in[5];

  float* out = (float*)d_out;
  float* ws  = (float*)d_ws;                  // needs 2*128*128 floats = 128 KB

  const long N = (long)in_sizes[1] / PDIM;    // B*S = 524288

  float* out_Gii = out;                                 // [P,P]
  float* out_Gti = out + (long)PDIM * PDIM;             // [N,P]
  float* out_gt  = out_Gti + N * (long)PDIM;            // [N]
  float* out_obj = out_gt + N;                          // scalar

  setup_kernel<<<1, 1024, 0, stream>>>(K_ii, V, gsc, dof, ws, out_Gii, out_obj);

  int nBlocks = (int)(N / 128);               // 8 waves x 16 rows per block
  fused_gram_kernel<<<nBlocks, 256, 0, stream>>>(K_ti, K_t, ws, out_Gti, out_gt);
}